// MakeCutouts_52742198395514
// MI455X (gfx1250) — compile-verified
//
#include <hip/hip_runtime.h>
#include <stdint.h>

// MakeCutouts for MI455X (gfx1250, wave32).
// Reference constants: B=4, C=3, IMG=512, CUTN=64, CUT_SIZE=224.
#define IMG   512
#define SOUT  224
#define NPLN  12                    // B*C planes
#define NTOT  (NPLN * IMG * IMG)    // total input elements
#define ROWF  IMG                   // floats staged per row (max crop width)
#define BLK   256                   // 8 waves of 32

// CDNA5 async global -> LDS copy (ASYNCcnt-tracked).
// VDST = per-lane LDS byte address (hardware adds the wave's LDS base; the
// low 32 bits of a generic pointer to __shared__ are exactly that offset).
// VADDR = 64-bit per-lane global address, SADDR=off (GV mode).
__device__ __forceinline__ void async_ld_b32(uint32_t lds_addr, const float* gptr) {
    asm volatile("global_load_async_to_lds_b32 %0, %1, off"
                 :: "v"(lds_addr), "v"((uint64_t)(uintptr_t)gptr)
                 : "memory");
}

__device__ __forceinline__ void wait_async_le4() {
    asm volatile("s_wait_asynccnt 0x4" ::: "memory");
}
__device__ __forceinline__ void wait_async_0() {
    asm volatile("s_wait_asynccnt 0x0" ::: "memory");
}

__global__ __launch_bounds__(BLK)
void make_cutouts_kernel(const float* __restrict__ x,
                         const int*   __restrict__ sizes,
                         const int*   __restrict__ offx,
                         const int*   __restrict__ offy,
                         float*       __restrict__ out)
{
    // Double-buffered staging: 2 buffers x (2 rows x 512 floats) = 8 KB LDS.
    __shared__ float lds[2][2 * ROWF];

    const int y = blockIdx.x;     // output row 0..223
    const int n = blockIdx.y;     // cutout    0..63
    const int t = threadIdx.x;    // 0..255

    // Block-uniform cutout params (scalar loads).
    const int   s  = sizes[n];
    const int   ox = offx[n];
    const int   oy = offy[n];
    const float sf = (float)s;
    const int   smax = s - 1;

    // Vertical mapping (uniform across block): src = max(s/S*(y+0.5)-0.5, 0)
    float srcy = fmaxf(sf * (1.0f / SOUT) * ((float)y + 0.5f) - 0.5f, 0.0f);
    int   i0y  = (int)srcy;                 // floor (srcy >= 0)
    float wy   = srcy - (float)i0y;         // frac before clamp (matches ref)
    if (i0y > smax) i0y = smax;
    int   i1y  = i0y + 1; if (i1y > smax) i1y = smax;

    // Horizontal mapping (per thread, reused for all 12 planes).
    float srcx = fmaxf(sf * (1.0f / SOUT) * ((float)t + 0.5f) - 0.5f, 0.0f);
    int   i0x  = (int)srcx;
    float wx   = srcx - (float)i0x;
    if (i0x > smax) i0x = smax;
    int   i1x  = i0x + 1; if (i1x > smax) i1x = smax;

    // Element offsets (within one plane) of the two source rows' crop start.
    const int row0 = (oy + i0y) * IMG + ox;
    const int row1 = (oy + i1y) * IMG + ox;

    const uint32_t lb0 = (uint32_t)(uintptr_t)&lds[0][0];
    const uint32_t lb1 = (uint32_t)(uintptr_t)&lds[1][0];

    // Stage plane p's two source rows (always 512 floats/row, clamped so
    // EXEC stays full => statically 4 async issues per wave per stage).
    auto stage = [&](int p, int sel) {
        const int      pb = p * (IMG * IMG);
        const int      j0 = t, j1 = t + BLK;
        const uint32_t lb = sel ? lb1 : lb0;
        int g00 = min(pb + row0 + j0, NTOT - 1);
        int g01 = min(pb + row0 + j1, NTOT - 1);
        int g10 = min(pb + row1 + j0, NTOT - 1);
        int g11 = min(pb + row1 + j1, NTOT - 1);
        async_ld_b32(lb + 4u * (uint32_t)j0,          x + g00);
        async_ld_b32(lb + 4u * (uint32_t)j1,          x + g01);
        async_ld_b32(lb + 4u * (uint32_t)(ROWF + j0), x + g10);
        async_ld_b32(lb + 4u * (uint32_t)(ROWF + j1), x + g11);
    };

    stage(0, 0);   // prologue: fill buffer 0 with plane 0

    const size_t outRow = (size_t)n * NPLN * (SOUT * SOUT) + (size_t)y * SOUT;

    for (int p = 0; p < NPLN; ++p) {
        const int sel = p & 1;
        if (p + 1 < NPLN) {
            stage(p + 1, sel ^ 1);   // overlap next plane's copy with compute
            wait_async_le4();        // in-order retire => stage(p) complete
        } else {
            wait_async_0();
        }
        __syncthreads();             // stage(p) visible to all waves

        if (t < SOUT) {
            const float* b0 = &lds[sel][0];
            const float* b1 = &lds[sel][ROWF];
            float v00 = b0[i0x], v01 = b0[i1x];
            float v10 = b1[i0x], v11 = b1[i1x];
            float top = v00 + wx * (v01 - v00);
            float bot = v10 + wx * (v11 - v10);
            // Streaming 154 MB output: non-temporal so it doesn't evict the
            // L2-resident 12.6 MB input.
            __builtin_nontemporal_store(top + wy * (bot - top),
                                        out + outRow + (size_t)p * (SOUT * SOUT) + t);
        }
        __syncthreads();             // reads of lds[sel] done before reuse at p+2
    }
}

extern "C" void kernel_launch(void* const* d_in, const int* in_sizes, int n_in,
                              void* d_out, int out_size, void* d_ws, size_t ws_size,
                              hipStream_t stream) {
    const float* x     = (const float*)d_in[0];
    const int*   sizes = (const int*)d_in[1];
    const int*   offx  = (const int*)d_in[2];
    const int*   offy  = (const int*)d_in[3];
    float*       out   = (float*)d_out;

    const int N = in_sizes[1];            // CUTN (64)
    dim3 grid(SOUT, N);                   // one block per (output row, cutout)
    make_cutouts_kernel<<<grid, BLK, 0, stream>>>(x, sizes, offx, offy, out);
    (void)n_in; (void)out_size; (void)d_ws; (void)ws_size;
}